// GINLayer_15272903705023
// MI455X (gfx1250) — compile-verified
//
#include <hip/hip_runtime.h>

#define D_FEAT   128
#define N_NODES  100000
#define ROWS     32          // rows per workgroup; 100000/32 = 3125 exact
#define A_STRIDE 132         // padded LDS row stride -> conflict-free column reads

typedef __attribute__((ext_vector_type(2))) float v2f;
typedef __attribute__((ext_vector_type(8))) float v8f;

// LDS layout (floats): sW[16384] (reused for W1 then W2) | sA[32*132] | sH[32*132]
#define SMEM_FLOATS (16384 + ROWS * A_STRIDE + ROWS * A_STRIDE)
#define SMEM_BYTES  (SMEM_FLOATS * 4)

// ---------------- Kernel 1: scatter-add x[col] into agg[row] ----------------
// One wave per edge (32 lanes x 4 features = 128). Edge id is forced into an
// SGPR via readfirstlane so the int64 index loads lower to scalar s_load_b64.
__global__ __launch_bounds__(256)
void gin_scatter(const float* __restrict__ x, const long long* __restrict__ ei,
                 float* __restrict__ agg, long long E) {
    int w = __builtin_amdgcn_readfirstlane((int)(threadIdx.x >> 5));
    long long e = (long long)blockIdx.x * 8 + w;
    if (e >= E) return;                      // wave-uniform branch
    long long row = ei[e];                   // edge_index[0][e]  (SMEM)
    long long col = ei[E + e];               // edge_index[1][e]  (SMEM)
    int d4 = (int)(threadIdx.x & 31) * 4;
    const float4 v = *reinterpret_cast<const float4*>(x + col * D_FEAT + d4);
    float* dst = agg + row * D_FEAT + d4;
    atomicAdd(dst + 0, v.x);
    atomicAdd(dst + 1, v.y);
    atomicAdd(dst + 2, v.z);
    atomicAdd(dst + 3, v.w);
}

// ---------------- Kernel 2: fused MLP via V_WMMA_F32_16X16X4_F32 ----------------
// One workgroup (8 waves) per 32-row tile. A = (1+eps)*x + agg computed while
// staging. Wave w owns output columns [16w,16w+16) for both 16-row M-tiles;
// B fragments are shared by two independent accumulator chains.
__global__ __launch_bounds__(256)
void gin_mlp(const float* __restrict__ x, const float* __restrict__ agg,
             const float* __restrict__ eps,
             const float* __restrict__ W1, const float* __restrict__ b1,
             const float* __restrict__ W2, const float* __restrict__ b2,
             float* __restrict__ out) {
    extern __shared__ float smem[];
    float* sW = smem;                     // 16384 floats, W1 then W2
    float* sA = smem + 16384;             // 32 x 132
    float* sH = sA + ROWS * A_STRIDE;     // 32 x 132

    const int tid   = threadIdx.x;
    const int wave  = tid >> 5;
    const int lane  = tid & 31;
    const int m16   = lane & 15;          // M (A frag) / N (B,C,D frags)
    const int khalf = lane >> 4;          // selects K pair {0,1} vs {2,3}
    const int row0  = blockIdx.x * ROWS;
    const int n     = wave * 16 + m16;    // this lane's output column

    // Stage W1 into LDS: 256 threads x 16 iters x float4.
#pragma unroll
    for (int i = 0; i < 16; ++i) {
        int off = tid * 4 + i * 1024;
        *reinterpret_cast<float4*>(sW + off) = *reinterpret_cast<const float4*>(W1 + off);
    }
    // Stage A tile: A = (1+eps)*x + agg. Thread t -> row t>>3, cols (t&7)*16..+15.
    {
        const float scale = 1.0f + eps[0];
        int r = tid >> 3, c = (tid & 7) * 16;
        const float* xs = x   + (long long)(row0 + r) * D_FEAT + c;
        const float* as = agg + (long long)(row0 + r) * D_FEAT + c;
        float* dst = sA + r * A_STRIDE + c;
#pragma unroll
        for (int q = 0; q < 4; ++q) {
            const float4 xv = *reinterpret_cast<const float4*>(xs + q * 4);
            const float4 av = *reinterpret_cast<const float4*>(as + q * 4);
            float4 o;
            o.x = fmaf(scale, xv.x, av.x);
            o.y = fmaf(scale, xv.y, av.y);
            o.z = fmaf(scale, xv.z, av.z);
            o.w = fmaf(scale, xv.w, av.w);
            *reinterpret_cast<float4*>(dst + q * 4) = o;
        }
    }
    __syncthreads();

    // ---- GEMM1: h = relu(A @ W1 + b1), two 16-row M-tiles per wave ----
    v8f acc0 = {}, acc1 = {};
#pragma unroll 4
    for (int kk = 0; kk < 32; ++kk) {
        int k0 = kk * 4 + khalf * 2;
        v2f b, a0, a1;
        b.x  = sW[k0 * D_FEAT + n];             // W1[k0][n]
        b.y  = sW[(k0 + 1) * D_FEAT + n];       // W1[k0+1][n]
        a0.x = sA[m16 * A_STRIDE + k0];         // M-tile 0 (rows 0..15)
        a0.y = sA[m16 * A_STRIDE + k0 + 1];
        a1.x = sA[(16 + m16) * A_STRIDE + k0];  // M-tile 1 (rows 16..31)
        a1.y = sA[(16 + m16) * A_STRIDE + k0 + 1];
        acc0 = __builtin_amdgcn_wmma_f32_16x16x4_f32(false, a0, false, b, (short)0, acc0, false, false);
        acc1 = __builtin_amdgcn_wmma_f32_16x16x4_f32(false, a1, false, b, (short)0, acc1, false, false);
    }
    {
        float bias1 = b1[n];
#pragma unroll
        for (int r = 0; r < 8; ++r) {
            int m = khalf * 8 + r;              // C/D layout: VGPR r -> row khalf*8+r
            float v0 = acc0[r] + bias1;
            float v1 = acc1[r] + bias1;
            sH[m * A_STRIDE + n]        = v0 > 0.0f ? v0 : 0.0f;
            sH[(16 + m) * A_STRIDE + n] = v1 > 0.0f ? v1 : 0.0f;
        }
    }
    __syncthreads();   // all waves done reading W1 and writing sH

    // Stage W2 into the same LDS buffer.
#pragma unroll
    for (int i = 0; i < 16; ++i) {
        int off = tid * 4 + i * 1024;
        *reinterpret_cast<float4*>(sW + off) = *reinterpret_cast<const float4*>(W2 + off);
    }
    __syncthreads();

    // ---- GEMM2: out = h @ W2 + b2 ----
    v8f acc2 = {}, acc3 = {};
#pragma unroll 4
    for (int kk = 0; kk < 32; ++kk) {
        int k0 = kk * 4 + khalf * 2;
        v2f b, a0, a1;
        b.x  = sW[k0 * D_FEAT + n];
        b.y  = sW[(k0 + 1) * D_FEAT + n];
        a0.x = sH[m16 * A_STRIDE + k0];
        a0.y = sH[m16 * A_STRIDE + k0 + 1];
        a1.x = sH[(16 + m16) * A_STRIDE + k0];
        a1.y = sH[(16 + m16) * A_STRIDE + k0 + 1];
        acc2 = __builtin_amdgcn_wmma_f32_16x16x4_f32(false, a0, false, b, (short)0, acc2, false, false);
        acc3 = __builtin_amdgcn_wmma_f32_16x16x4_f32(false, a1, false, b, (short)0, acc3, false, false);
    }
    {
        float bias2 = b2[n];
#pragma unroll
        for (int r = 0; r < 8; ++r) {
            int m = khalf * 8 + r;
            out[(long long)(row0 + m) * D_FEAT + n]      = acc2[r] + bias2;
            out[(long long)(row0 + 16 + m) * D_FEAT + n] = acc3[r] + bias2;
        }
    }
}

// ---------------- Host launcher ----------------
extern "C" void kernel_launch(void* const* d_in, const int* in_sizes, int n_in,
                              void* d_out, int out_size, void* d_ws, size_t ws_size,
                              hipStream_t stream) {
    const float*     x   = (const float*)d_in[0];
    const long long* ei  = (const long long*)d_in[1];   // int64 [2, E]
    // d_in[2] = num_nodes (scalar, unused; N is compile-time)
    const float*     eps = (const float*)d_in[3];
    const float*     W1  = (const float*)d_in[4];
    const float*     b1  = (const float*)d_in[5];
    const float*     W2  = (const float*)d_in[6];
    const float*     b2  = (const float*)d_in[7];
    float* out = (float*)d_out;
    float* agg = (float*)d_ws;                          // N*128 f32 = 51.2 MB scratch

    const long long E = (long long)in_sizes[1] / 2;

    // Allow ~97 KB dynamic LDS for the MLP kernel (deterministic, no stream work).
    hipFuncSetAttribute((const void*)gin_mlp,
                        hipFuncAttributeMaxDynamicSharedMemorySize, SMEM_BYTES);

    // 1) agg = 0 (graph-capturable async memset; replaces a full read of x)
    hipMemsetAsync(agg, 0, (size_t)N_NODES * D_FEAT * sizeof(float), stream);

    // 2) scatter-add edges: one wave per edge, 8 edges per 256-thread block
    gin_scatter<<<(int)((E + 7) / 8), 256, 0, stream>>>(x, ei, agg, E);

    // 3) fused 2-layer MLP with f32 WMMA; 100000/32 = 3125 tiles exactly
    gin_mlp<<<N_NODES / ROWS, 256, SMEM_BYTES, stream>>>(x, agg, eps, W1, b1, W2, b2, out);
}